// SoftminVectorQuantizationLayer_41772851921060
// MI455X (gfx1250) — compile-verified
//
#include <hip/hip_runtime.h>

typedef __attribute__((ext_vector_type(2))) float v2f;
typedef __attribute__((ext_vector_type(8))) float v8f;

#define B_SZ 8192
#define G_SZ 4
#define K_SZ 512
#define D_SZ 512
#define BGD  (B_SZ * G_SZ * D_SZ)   // 16777216
#define BETA 0.25f

// workspace layout (in floats); total < 0.5 MB
#define WS_PSQ  0                          // G*K   = 2048
#define WS_XSQ  (WS_PSQ + G_SZ * K_SZ)     // B*G   = 32768
#define WS_RMAX (WS_XSQ + B_SZ * G_SZ)     // G*B   = 32768
#define WS_RSUM (WS_RMAX + G_SZ * B_SZ)    // G*B   = 32768
#define WS_PART (WS_RSUM + G_SZ * B_SZ)    // 512 block partials
#define NPART   512

// ---------------------------------------------------------------------------
// Kernel 1: row sums of squares for protos (p_sq) and latents (x_sq).
// One wave per row of 512 f32; rows = G*K (protos) then B*G (latents).
// ---------------------------------------------------------------------------
__global__ void k_rowsq(const float* __restrict__ latents,
                        const float* __restrict__ protos,
                        float* __restrict__ ws) {
  const int wave = blockIdx.x * (blockDim.x >> 5) + (threadIdx.x >> 5);
  const int lane = threadIdx.x & 31;
  const int nProtoRows = G_SZ * K_SZ;
  const float* base = (wave < nProtoRows)
                          ? (protos + (size_t)wave * D_SZ)
                          : (latents + (size_t)(wave - nProtoRows) * D_SZ);
  float acc = 0.f;
#pragma unroll
  for (int i = 0; i < 4; ++i) {
    const float4 v = ((const float4*)base)[lane + 32 * i];
    acc += v.x * v.x + v.y * v.y + v.z * v.z + v.w * v.w;
  }
#pragma unroll
  for (int off = 16; off; off >>= 1) acc += __shfl_xor(acc, off, 32);
  if (lane == 0) {
    if (wave < nProtoRows) ws[WS_PSQ + wave] = acc;
    else                   ws[WS_XSQ + (wave - nProtoRows)] = acc;
  }
}

// ---------------------------------------------------------------------------
// Kernel 2: GEMM1  cross[g,b,k] = latents[b,g,:]·protos[g,k,:]  via
// V_WMMA_F32_16X16X4_F32, fused epilogue  dists = x_sq - 2*cross + p_sq.
// Block = 8 waves = 128(b) x 256(k) tile; wave = 64x64 (4x4 WMMA subtiles).
// ---------------------------------------------------------------------------
__global__ __launch_bounds__(256) void k_gemm1(
    const float* __restrict__ latents, const float* __restrict__ protos,
    const float* __restrict__ ws, float* __restrict__ dists) {
  const int tid  = threadIdx.x;
  const int lane = tid & 31;
  const int w    = tid >> 5;
  const int wm   = w & 1;        // 2 waves along M
  const int wn   = w >> 1;       // 4 waves along N
  const int g    = blockIdx.z;
  const int bm0  = blockIdx.x * 128 + wm * 64;
  const int kn0  = blockIdx.y * 256 + wn * 64;
  const int h    = lane >> 4;    // lane half (K-pair select)
  const int r    = lane & 15;    // row/col within fragment

  // A-frag lane(h, m=r): A[m][2h], A[m][2h+1]  -> contiguous float2
  const float* aP[4];
#pragma unroll
  for (int i = 0; i < 4; ++i)
    aP[i] = latents + (size_t)(bm0 + 16 * i + r) * (G_SZ * D_SZ) + g * D_SZ + 2 * h;
  // B-frag lane(h, n=r): P[kn0+n][2h], P[kn0+n][2h+1] -> contiguous float2
  const float* bP[4];
#pragma unroll
  for (int j = 0; j < 4; ++j)
    bP[j] = protos + (size_t)g * K_SZ * D_SZ + (size_t)(kn0 + 16 * j + r) * D_SZ + 2 * h;

  v8f acc[4][4] = {};
#pragma unroll 1
  for (int d0 = 0; d0 < D_SZ; d0 += 4) {
    v2f a[4], b[4];
#pragma unroll
    for (int i = 0; i < 4; ++i) a[i] = *(const v2f*)(aP[i] + d0);
#pragma unroll
    for (int j = 0; j < 4; ++j) b[j] = *(const v2f*)(bP[j] + d0);
#pragma unroll
    for (int i = 0; i < 4; ++i)
#pragma unroll
      for (int j = 0; j < 4; ++j)
        acc[i][j] = __builtin_amdgcn_wmma_f32_16x16x4_f32(
            false, a[i], false, b[j], (short)0, acc[i][j], false, false);
  }

  // Epilogue: dists[g][b][k] = x_sq[b][g] - 2*cross + p_sq[g][k]
  const float* psq = ws + WS_PSQ + g * K_SZ;
  const float* xsq = ws + WS_XSQ;
  float* dg = dists + (size_t)g * B_SZ * K_SZ;
#pragma unroll
  for (int i = 0; i < 4; ++i) {
    float xs[8];
#pragma unroll
    for (int v = 0; v < 8; ++v)
      xs[v] = xsq[(bm0 + 16 * i + 8 * h + v) * G_SZ + g];
#pragma unroll
    for (int j = 0; j < 4; ++j) {
      const int k = kn0 + 16 * j + r;
      const float ps = psq[k];
#pragma unroll
      for (int v = 0; v < 8; ++v) {
        const int b = bm0 + 16 * i + 8 * h + v;
        dg[(size_t)b * K_SZ + k] = xs[v] - 2.0f * acc[i][j][v] + ps;
      }
    }
  }
}

// ---------------------------------------------------------------------------
// Kernel 3: per-(g,b) softmin stats: m = max(-t*d), Z = sum exp(-t*d - m).
// One wave per row; 32-bit loads only (dists base is 4-byte-odd aligned).
// ---------------------------------------------------------------------------
__global__ void k_stats(const float* __restrict__ dists,
                        const float* __restrict__ temp_p,
                        float* __restrict__ ws) {
  const int wave = blockIdx.x * (blockDim.x >> 5) + (threadIdx.x >> 5);
  const int lane = threadIdx.x & 31;
  const float t = temp_p[0];
  const float* row = dists + (size_t)wave * K_SZ;
  float s[16];
  float m = -3.4e38f;
#pragma unroll
  for (int i = 0; i < 16; ++i) {
    s[i] = -t * row[lane + 32 * i];
    m = fmaxf(m, s[i]);
  }
#pragma unroll
  for (int off = 16; off; off >>= 1) m = fmaxf(m, __shfl_xor(m, off, 32));
  float z = 0.f;
#pragma unroll
  for (int i = 0; i < 16; ++i) z += __expf(s[i] - m);
#pragma unroll
  for (int off = 16; off; off >>= 1) z += __shfl_xor(z, off, 32);
  if (lane == 0) { ws[WS_RMAX + wave] = m; ws[WS_RSUM + wave] = z; }
}

// ---------------------------------------------------------------------------
// Kernel 4: GEMM2  quantized[b,g,d] = softmin_weights[g,b,:]·protos[g,:,d]
// Weights materialized on the fly: w = exp(-t*d - m) / Z.
// Same tiling as GEMM1. Epilogue writes quantized (== quantized_st forward)
// and accumulates sum(diff^2) deterministically into per-block partials.
// ---------------------------------------------------------------------------
__global__ __launch_bounds__(256) void k_gemm2(
    const float* __restrict__ latents, const float* __restrict__ protos,
    const float* __restrict__ dists, const float* __restrict__ ws,
    const float* __restrict__ temp_p, float* __restrict__ qout,
    float* __restrict__ part) {
  __shared__ float red[8];
  const int tid  = threadIdx.x;
  const int lane = tid & 31;
  const int w    = tid >> 5;
  const int wm   = w & 1;
  const int wn   = w >> 1;
  const int g    = blockIdx.z;
  const int bm0  = blockIdx.x * 128 + wm * 64;
  const int dn0  = blockIdx.y * 256 + wn * 64;
  const int h    = lane >> 4;
  const int r    = lane & 15;
  const float t  = temp_p[0];

  const float* distsG = dists + (size_t)g * B_SZ * K_SZ;
  const float* aP[4];
  float mrow[4], izrow[4];
#pragma unroll
  for (int i = 0; i < 4; ++i) {
    const int b = bm0 + 16 * i + r;
    aP[i]    = distsG + (size_t)b * K_SZ + 2 * h;
    mrow[i]  = ws[WS_RMAX + g * B_SZ + b];
    izrow[i] = 1.0f / ws[WS_RSUM + g * B_SZ + b];
  }
  // B-frag lane(h, n=r): P[k0+2h+v][dn0+n]  (stride-D scalar loads)
  const float* bP[4];
#pragma unroll
  for (int j = 0; j < 4; ++j)
    bP[j] = protos + (size_t)g * K_SZ * D_SZ + (size_t)(2 * h) * D_SZ + (dn0 + 16 * j + r);

  v8f acc[4][4] = {};
#pragma unroll 1
  for (int k0 = 0; k0 < K_SZ; k0 += 4) {
    v2f a[4], b[4];
#pragma unroll
    for (int i = 0; i < 4; ++i) {
      const float d0v = aP[i][k0];
      const float d1v = aP[i][k0 + 1];
      a[i].x = __expf(fmaf(-t, d0v, -mrow[i])) * izrow[i];
      a[i].y = __expf(fmaf(-t, d1v, -mrow[i])) * izrow[i];
    }
#pragma unroll
    for (int j = 0; j < 4; ++j) {
      b[j].x = bP[j][(size_t)k0 * D_SZ];
      b[j].y = bP[j][(size_t)(k0 + 1) * D_SZ];
    }
#pragma unroll
    for (int i = 0; i < 4; ++i)
#pragma unroll
      for (int j = 0; j < 4; ++j)
        acc[i][j] = __builtin_amdgcn_wmma_f32_16x16x4_f32(
            false, a[i], false, b[j], (short)0, acc[i][j], false, false);
  }

  // Epilogue: write quantized, accumulate diff^2
  float lsum = 0.f;
#pragma unroll
  for (int i = 0; i < 4; ++i)
#pragma unroll
    for (int j = 0; j < 4; ++j) {
      const int d = dn0 + 16 * j + r;
#pragma unroll
      for (int v = 0; v < 8; ++v) {
        const int b = bm0 + 16 * i + 8 * h + v;
        const size_t idx = (size_t)b * (G_SZ * D_SZ) + g * D_SZ + d;
        const float q = acc[i][j][v];
        qout[idx] = q;
        const float diff = q - latents[idx];
        lsum = fmaf(diff, diff, lsum);
      }
    }
#pragma unroll
  for (int off = 16; off; off >>= 1) lsum += __shfl_xor(lsum, off, 32);
  if (lane == 0) red[w] = lsum;
  __syncthreads();
  if (tid == 0) {
    float s = 0.f;
#pragma unroll
    for (int i = 0; i < 8; ++i) s += red[i];
    part[blockIdx.x + 64 * (blockIdx.y + 2 * blockIdx.z)] = s;
  }
}

// ---------------------------------------------------------------------------
// Kernel 5: reduce NPART block partials -> vq_loss = 1.25 * total / (G*B*D)
// ---------------------------------------------------------------------------
__global__ void k_loss(const float* __restrict__ part, float* __restrict__ loss) {
  __shared__ float red[256];
  const int t = threadIdx.x;
  red[t] = part[t] + part[t + 256];
  __syncthreads();
  for (int s = 128; s; s >>= 1) {
    if (t < s) red[t] += red[t + s];
    __syncthreads();
  }
  if (t == 0)
    loss[0] = (1.0f + BETA) * red[0] / (float)((size_t)G_SZ * B_SZ * D_SZ);
}

// ---------------------------------------------------------------------------
extern "C" void kernel_launch(void* const* d_in, const int* in_sizes, int n_in,
                              void* d_out, int out_size, void* d_ws,
                              size_t ws_size, hipStream_t stream) {
  const float* latents = (const float*)d_in[0];
  const float* protos  = (const float*)d_in[1];
  const float* temp    = (const float*)d_in[2];
  float* out   = (float*)d_out;
  float* q     = out;             // [B,G,D]
  float* loss  = out + BGD;       // scalar
  float* dists = out + BGD + 1;   // [G,B,K]
  float* ws    = (float*)d_ws;

  {
    const int rows = G_SZ * K_SZ + B_SZ * G_SZ;   // 34816, divisible by 8
    k_rowsq<<<rows / 8, 256, 0, stream>>>(latents, protos, ws);
  }
  {
    dim3 grid(B_SZ / 128, K_SZ / 256, G_SZ);      // 64 x 2 x 4
    k_gemm1<<<grid, 256, 0, stream>>>(latents, protos, ws, dists);
  }
  {
    const int rows = G_SZ * B_SZ;                 // 32768, divisible by 8
    k_stats<<<rows / 8, 256, 0, stream>>>(dists, temp, ws);
  }
  {
    dim3 grid(B_SZ / 128, D_SZ / 256, G_SZ);      // 64 x 2 x 4 = 512 blocks
    k_gemm2<<<grid, 256, 0, stream>>>(latents, protos, dists, ws, temp, q,
                                      ws + WS_PART);
  }
  k_loss<<<1, 256, 0, stream>>>(ws + WS_PART, loss);
}